// HyperParamSelfAttn_55259049230542
// MI455X (gfx1250) — compile-verified
//
#include <hip/hip_runtime.h>

#define ENC 1024
#define HD  4096
#define OUTROWS 49152      // DIM * BOT
#define KC 64              // K-chunk staged in LDS per iteration
#define RPB 128            // rows of W2 per block (8 waves * 16 rows)
#define LSTRIDE 68         // 64 cols + 4 floats pad: float4-aligned, bank-conflict free
#define BUFSZ (RPB * LSTRIDE)

typedef __attribute__((ext_vector_type(2))) float v2f;
typedef __attribute__((ext_vector_type(4))) float v4f;
typedef __attribute__((ext_vector_type(8))) float v8f;
typedef __attribute__((ext_vector_type(4))) int   v4i;

#if defined(__has_builtin)
#if __has_builtin(__builtin_amdgcn_global_load_async_to_lds_b128) && \
    __has_builtin(__builtin_amdgcn_s_wait_asynccnt)
#define USE_ASYNC_LDS 1
#endif
#endif

static __device__ __forceinline__ float wave_sum(float v) {
  #pragma unroll
  for (int off = 16; off > 0; off >>= 1) v += __shfl_xor(v, off, 32);
  return v;
}

// ---------------- generic one-row-per-wave GEMV ----------------
static __device__ __forceinline__ void gemv_row(const float* __restrict__ W,
                                                const float* __restrict__ x,
                                                const float* __restrict__ b,
                                                float* __restrict__ y,
                                                int cols, int do_relu) {
  const int wave = threadIdx.x >> 5;
  const int lane = threadIdx.x & 31;
  const int row  = blockIdx.x * 8 + wave;
  const float* Wr = W + (size_t)row * cols;
  float acc = 0.f;
  const int iters = cols >> 7;               // cols / (32 lanes * 4 floats)
  for (int i = 0; i < iters; ++i) {
    int c = (lane + (i << 5)) << 2;
    v4f w  = *(const v4f*)(Wr + c);
    v4f xv = *(const v4f*)(x + c);
    acc += w.x * xv.x + w.y * xv.y + w.z * xv.z + w.w * xv.w;
  }
  acc = wave_sum(acc);
  if (lane == 0) {
    float r = acc + b[row];
    y[row] = do_relu ? fmaxf(r, 0.f) : r;
  }
}

__global__ void k_h(const float* __restrict__ W1, const float* __restrict__ x,
                    const float* __restrict__ b1, float* __restrict__ h) {
  gemv_row(W1, x, b1, h, ENC, 1);
}

__global__ void k_qkv(const float* __restrict__ Wq, const float* __restrict__ bq,
                      const float* __restrict__ Wk, const float* __restrict__ bk,
                      const float* __restrict__ Wv, const float* __restrict__ bv,
                      const float* __restrict__ h,
                      float* __restrict__ Q, float* __restrict__ K, float* __restrict__ V) {
  const float* W; const float* b; float* y;
  if (blockIdx.y == 0)      { W = Wq; b = bq; y = Q; }
  else if (blockIdx.y == 1) { W = Wk; b = bk; y = K; }
  else                      { W = Wv; b = bv; y = V; }
  gemv_row(W, h, b, y, HD, 0);
}

// ---------------- K min/max (for softmax row max of rank-1 scores) -----------
__global__ void k_minmax(const float* __restrict__ K, float* __restrict__ mm) {
  __shared__ float smax[256], smin[256];
  int t = threadIdx.x;
  float mx = -1e30f, mn = 1e30f;
  for (int i = t; i < HD; i += 256) { float v = K[i]; mx = fmaxf(mx, v); mn = fminf(mn, v); }
  smax[t] = mx; smin[t] = mn;
  __syncthreads();
  for (int s = 128; s > 0; s >>= 1) {
    if (t < s) { smax[t] = fmaxf(smax[t], smax[t + s]); smin[t] = fminf(smin[t], smin[t + s]); }
    __syncthreads();
  }
  if (t == 0) { mm[0] = smax[0]; mm[1] = smin[0]; }
}

// ---------------- rank-1 attention: out[i] = softmax(q_i*K) @ V + h[i] -------
__global__ void k_attn(const float* __restrict__ Q, const float* __restrict__ K,
                       const float* __restrict__ V, const float* __restrict__ h,
                       const float* __restrict__ mm, float* __restrict__ out) {
  __shared__ float sK[HD];
  __shared__ float sV[HD];
  int t = threadIdx.x;
  for (int i = t; i < HD; i += 256) { sK[i] = K[i]; sV[i] = V[i]; }
  __syncthreads();
  int i = blockIdx.x * 256 + t;
  float q = Q[i] * 0.015625f;                     // 1/sqrt(4096)
  float m = (q >= 0.f) ? q * mm[0] : q * mm[1];   // row max of q*K
  float S = 0.f, T = 0.f;
  for (int j = 0; j < HD; ++j) {
    float e = __expf(fmaf(q, sK[j], -m));         // v_exp_f32
    S += e;
    T = fmaf(e, sV[j], T);
  }
  out[i] = T / S + h[i];
}

// ---------------- dominant GEMV: y = W2 @ out + b2, WMMA f32 16x16x4 ---------
// Double-buffered: async global->LDS for chunk k+1 overlaps WMMA on chunk k.
__global__ void __launch_bounds__(256)
k_w2(const float* __restrict__ W2, const float* __restrict__ v,
     const float* __restrict__ b2, float* __restrict__ y) {
  __shared__ float tile[2 * BUFSZ];               // 69632 B (of 320 KB/WGP)
  const int tid  = threadIdx.x;
  const int wave = tid >> 5;
  const int lane = tid & 31;
  const int half = lane >> 4;                     // 0: K pair {0,1}; 1: {2,3}
  const int m    = lane & 15;                     // A-matrix row within 16-row tile
  const size_t rowBase = (size_t)blockIdx.x * RPB;

  // stage W2[rowBase..rowBase+127][k0..k0+63] into LDS buffer `buf` (coalesced)
  auto stage = [&](int k0, int buf) {
    #pragma unroll
    for (int i = 0; i < (RPB * KC) / (256 * 4); ++i) {   // 8 float4s per thread
      int idx = tid + (i << 8);                  // float4 index 0..2047
      int r   = idx >> 4;                        // row 0..127
      int c4  = idx & 15;                        // float4 within row
      const float* gp = W2 + (rowBase + r) * (size_t)HD + k0 + (c4 << 2);
      float*       lp = &tile[buf * BUFSZ + r * LSTRIDE + (c4 << 2)];
#if USE_ASYNC_LDS
      typedef __attribute__((address_space(1))) v4i gv4i_t;   // global int4
      typedef __attribute__((address_space(3))) v4i lv4i_t;   // LDS int4
      __builtin_amdgcn_global_load_async_to_lds_b128((gv4i_t*)gp, (lv4i_t*)lp, 0, 0);
#else
      *(v4f*)lp = *(const v4f*)gp;
#endif
    }
  };

  v8f acc = {0.f, 0.f, 0.f, 0.f, 0.f, 0.f, 0.f, 0.f};

  stage(0, 0);
#if USE_ASYNC_LDS
  __builtin_amdgcn_s_wait_asynccnt(0);
#endif
  __syncthreads();

  for (int k0 = 0; k0 < HD; k0 += KC) {
    const int cur = (k0 / KC) & 1;

    // prefetch next chunk into the other buffer while we compute this one
    if (k0 + KC < HD) stage(k0 + KC, cur ^ 1);

    // preload all B fragments for this chunk (broadcast vector values; L0 hits,
    // issued as one clause so no global waits sit inside the WMMA chain)
    v2f bfr[KC / 4];
    #pragma unroll
    for (int s = 0; s < KC / 4; ++s)
      bfr[s] = *(const v2f*)(v + k0 + 4 * s + 2 * half);

    // 16 WMMA steps: wave accumulates its 16-row tile over KC columns
    const float* arow = &tile[cur * BUFSZ + (wave * 16 + m) * LSTRIDE + 2 * half];
    #pragma unroll
    for (int s = 0; s < KC / 4; ++s) {
      // A (16x4 f32): lane m, K = 4s + 2*half + {0,1}  (ISA 32-bit A layout)
      v2f a = *(const v2f*)(arow + 4 * s);
      acc = __builtin_amdgcn_wmma_f32_16x16x4_f32(false, a, false, bfr[s],
                                                  (short)0, acc, false, false);
    }

    // wait own async writes to the next buffer, then barrier:
    // - guarantees next buffer is fully written before anyone reads it
    // - guarantees everyone finished reading `cur` before it is re-staged
#if USE_ASYNC_LDS
    __builtin_amdgcn_s_wait_asynccnt(0);
#endif
    __syncthreads();
  }

  // D layout: VGPR i holds M=i (lanes 0-15) / M=i+8 (lanes 16-31); all N equal.
  if (m == 0) {
    size_t rb = rowBase + (size_t)wave * 16 + 8 * half;
    #pragma unroll
    for (int i = 0; i < 8; ++i) y[rb + i] = acc[i] + b2[rb + i];
  }
}

// ---------------- launch ----------------
extern "C" void kernel_launch(void* const* d_in, const int* in_sizes, int n_in,
                              void* d_out, int out_size, void* d_ws, size_t ws_size,
                              hipStream_t stream) {
  const float* x  = (const float*)d_in[0];
  const float* W1 = (const float*)d_in[1];
  const float* b1 = (const float*)d_in[2];
  const float* Wq = (const float*)d_in[3];
  const float* bq = (const float*)d_in[4];
  const float* Wk = (const float*)d_in[5];
  const float* bk = (const float*)d_in[6];
  const float* Wv = (const float*)d_in[7];
  const float* bv = (const float*)d_in[8];
  const float* W2 = (const float*)d_in[9];
  const float* b2 = (const float*)d_in[10];
  float* out = (float*)d_out;

  float* ws = (float*)d_ws;
  float* h  = ws;               // 4096
  float* Q  = ws + 4096;        // 4096
  float* K  = ws + 8192;        // 4096
  float* V  = ws + 12288;       // 4096
  float* ao = ws + 16384;       // 4096 (attention output + residual)
  float* mm = ws + 20480;       // 2

  k_h<<<dim3(HD / 8), dim3(256), 0, stream>>>(W1, x, b1, h);
  k_qkv<<<dim3(HD / 8, 3), dim3(256), 0, stream>>>(Wq, bq, Wk, bk, Wv, bv, h, Q, K, V);
  k_minmax<<<dim3(1), dim3(256), 0, stream>>>(K, mm);
  k_attn<<<dim3(HD / 256), dim3(256), 0, stream>>>(Q, K, V, h, mm, ao);
  k_w2<<<dim3(OUTROWS / RPB), dim3(256), 0, stream>>>(W2, ao, b2, out);
}